// PGNDecoder_77721728189098
// MI455X (gfx1250) — compile-verified
//
#include <hip/hip_runtime.h>
#include <math.h>

// ---------------------------------------------------------------------------
// PGN decoder step for MI455X (gfx1250, wave32, WMMA f32 16x16x4).
// Sizes: B=64, H=512, S=400, V=50000.
// HBM budget: out_W 205MB (streamed once, NT) + enc 52MB (L2-resident, RT).
// ---------------------------------------------------------------------------

#define B_   64
#define H_   512
#define S_   400
#define V_   50000
#define EXTW (V_ + S_)          // 50400

typedef float v2f __attribute__((ext_vector_type(2)));
typedef float v8f __attribute__((ext_vector_type(8)));

static __device__ __forceinline__ v8f wmma_f32(v2f a, v2f b, v8f c) {
  // V_WMMA_F32_16X16X4_F32 : D(16x16 f32) = A(16x4 f32) * B(4x16 f32) + C
  return __builtin_amdgcn_wmma_f32_16x16x4_f32(
      false, a, false, b, (short)0, c, false, false);
}

static __device__ __forceinline__ v8f v8zero() {
  v8f z = {0.f, 0.f, 0.f, 0.f, 0.f, 0.f, 0.f, 0.f};
  return z;
}

static __device__ __forceinline__ float sigmoidf_(float x) {
  return 1.0f / (1.0f + expf(-x));
}

// ---------------- workspace layout (floats) ----------------
#define GS_OFF   0                         // global_state (B,1024): [h | context]
#define GI_OFF   (GS_OFF + B_ * 1024)      // gi (B,1536)
#define GH_OFF   (GI_OFF + B_ * 1536)      // gh (B,1536)
#define X2_OFF   (GH_OFF + B_ * 1536)      // x2 = h@Ws (B,512)
#define PART_OFF (X2_OFF + B_ * H_)        // score partials (S*B, 8)
#define P_OFF    (PART_OFF + S_ * B_ * 8)  // p (B)

// ---------------- output layout (floats) ----------------
#define HID_OFF  (B_ * EXTW)               // 3,225,600
#define POUT_OFF (HID_OFF + B_ * H_)       // 3,258,368
#define ATTN_OFF (POUT_OFF + B_)           // 3,258,432  (S,B)

// ===========================================================================
// K1: gi = x @ W_ih^T + b_ih ; gh = h0 @ W_hh^T + b_hh   (WMMA)
// 96 N-tiles each (N=1536), M=64 (4 M-tiles per wave), K=512. 192 waves.
// ===========================================================================
__global__ void k_gru_gates_gemm(const float* __restrict__ x,
                                 const float* __restrict__ h0,
                                 const float* __restrict__ W_ih,
                                 const float* __restrict__ W_hh,
                                 const float* __restrict__ b_ih,
                                 const float* __restrict__ b_hh,
                                 float* __restrict__ gi,
                                 float* __restrict__ gh) {
  const int gwave = (blockIdx.x * blockDim.x + threadIdx.x) >> 5;
  const int lane  = threadIdx.x & 31;
  if (gwave >= 192) return;
  const bool is_gh = (gwave >= 96);
  const int  nt    = is_gh ? (gwave - 96) : gwave;

  const float* A    = is_gh ? h0 : x;       // (64,512)
  const float* W    = is_gh ? W_hh : W_ih;  // (1536,512) row-major
  const float* bias = is_gh ? b_hh : b_ih;
  float*       out  = is_gh ? gh : gi;      // (64,1536)

  const int half = lane >> 4;
  const int l16  = lane & 15;
  const int ncol = nt * 16 + l16;           // 0..1535 (weight row = output col)

  v8f acc[4];
#pragma unroll
  for (int mt = 0; mt < 4; ++mt) acc[mt] = v8zero();

  const float* wrow = W + (size_t)ncol * H_;
  for (int k = 0; k < H_; k += 4) {
    const int koff = k + half * 2;
    v2f bm = *(const v2f*)(wrow + koff);     // B[kk][n] = W[n][kk] (transposed)
#pragma unroll
    for (int mt = 0; mt < 4; ++mt) {
      const float* arow = A + (size_t)(mt * 16 + l16) * H_;
      v2f am = *(const v2f*)(arow + koff);
      acc[mt] = wmma_f32(am, bm, acc[mt]);
    }
  }

  const float bv = bias[ncol];
#pragma unroll
  for (int mt = 0; mt < 4; ++mt) {
#pragma unroll
    for (int i = 0; i < 8; ++i) {
      const int b = mt * 16 + i + half * 8;
      out[(size_t)b * 1536 + ncol] = acc[mt][i] + bv;
    }
  }
}

// ===========================================================================
// K2: combine gates -> h ; write h into gs[:,0:512] and d_out hidden.
// ===========================================================================
__global__ void k_gru_combine(const float* __restrict__ gi,
                              const float* __restrict__ gh,
                              const float* __restrict__ h0,
                              float* __restrict__ gs,
                              float* __restrict__ hid_out) {
  const int idx = blockIdx.x * blockDim.x + threadIdx.x;  // B*H = 32768
  if (idx >= B_ * H_) return;
  const int b = idx >> 9;
  const int j = idx & (H_ - 1);
  const size_t base = (size_t)b * 1536;
  const float ir = gi[base + j], iz = gi[base + H_ + j], in_ = gi[base + 2 * H_ + j];
  const float hr = gh[base + j], hz = gh[base + H_ + j], hn = gh[base + 2 * H_ + j];
  const float r = sigmoidf_(ir + hr);
  const float z = sigmoidf_(iz + hz);
  const float n = tanhf(in_ + r * hn);
  const float h = (1.0f - z) * n + z * h0[idx];
  gs[(size_t)b * 1024 + j] = h;
  hid_out[idx] = h;
}

// ===========================================================================
// K3: x2 = h @ Ws  (WMMA). M=64 (4 tiles), N=512 (32 tiles, 1 wave each), K=512.
// ===========================================================================
__global__ void k_x2_gemm(const float* __restrict__ gs,
                          const float* __restrict__ Ws,
                          float* __restrict__ x2) {
  const int gwave = (blockIdx.x * blockDim.x + threadIdx.x) >> 5;
  const int lane  = threadIdx.x & 31;
  if (gwave >= 32) return;
  const int half = lane >> 4;
  const int l16  = lane & 15;
  const int ncol = gwave * 16 + l16;

  v8f acc[4];
#pragma unroll
  for (int mt = 0; mt < 4; ++mt) acc[mt] = v8zero();

  for (int k = 0; k < H_; k += 4) {
    const int koff = k + half * 2;
    v2f bm;                                   // natural orientation, stride H_
    bm.x = Ws[(size_t)koff * H_ + ncol];
    bm.y = Ws[(size_t)(koff + 1) * H_ + ncol];
#pragma unroll
    for (int mt = 0; mt < 4; ++mt) {
      const float* arow = gs + (size_t)(mt * 16 + l16) * 1024;
      v2f am = *(const v2f*)(arow + koff);
      acc[mt] = wmma_f32(am, bm, acc[mt]);
    }
  }
#pragma unroll
  for (int mt = 0; mt < 4; ++mt) {
#pragma unroll
    for (int i = 0; i < 8; ++i) {
      const int b = mt * 16 + i + half * 8;
      x2[(size_t)b * H_ + ncol] = acc[mt][i];
    }
  }
}

// ===========================================================================
// K4: fused enc(25600x512) @ Wh(512x512) -> tanh(+x2+b_attn)*v -> row sums.
// Wave = (m-tile mt in [0,1600), n-group ntg in [0,8)); deterministic partials.
// enc stays RT so it remains L2-resident for K6 (52MB << 192MB L2).
// ===========================================================================
__global__ void k_attn_scores(const float* __restrict__ enc,
                              const float* __restrict__ Wh,
                              const float* __restrict__ x2,
                              const float* __restrict__ b_attn,
                              const float* __restrict__ vvec,
                              float* __restrict__ partials) {
  const int gwave = (blockIdx.x * blockDim.x + threadIdx.x) >> 5;  // 12800
  const int lane  = threadIdx.x & 31;
  if (gwave >= 1600 * 8) return;
  const int mt  = gwave >> 3;
  const int ntg = gwave & 7;
  const int half = lane >> 4;
  const int l16  = lane & 15;
  const int rowbase = mt * 16;

  v8f acc[4];
#pragma unroll
  for (int j = 0; j < 4; ++j) acc[j] = v8zero();

  for (int k = 0; k < H_; k += 4) {
    const int koff = k + half * 2;
    const float* arow = enc + (size_t)(rowbase + l16) * H_;
    v2f am = *(const v2f*)(arow + koff);
#pragma unroll
    for (int j = 0; j < 4; ++j) {
      const int ncol = (ntg * 4 + j) * 16 + l16;
      v2f bm;
      bm.x = Wh[(size_t)koff * H_ + ncol];
      bm.y = Wh[(size_t)(koff + 1) * H_ + ncol];
      acc[j] = wmma_f32(am, bm, acc[j]);
    }
  }

#pragma unroll
  for (int i = 0; i < 8; ++i) {
    const int gr = rowbase + i + half * 8;   // global (s*64+b) row
    const int b  = gr & 63;
    float s = 0.0f;
#pragma unroll
    for (int j = 0; j < 4; ++j) {
      const int ncol = (ntg * 4 + j) * 16 + l16;
      s += tanhf(acc[j][i] + x2[(size_t)b * H_ + ncol] + b_attn[ncol]) * vvec[ncol];
    }
#pragma unroll
    for (int m = 8; m >= 1; m >>= 1) s += __shfl_xor(s, m, 16);
    if (l16 == 0) partials[(size_t)gr * 8 + ntg] = s;
  }
}

// ===========================================================================
// K5: softmax over S per batch column b. One block per b.
// ===========================================================================
__global__ void k_softmax(const float* __restrict__ partials,
                          float* __restrict__ attn_out) {
  __shared__ float sc[S_];
  __shared__ float red[128];
  const int b   = blockIdx.x;
  const int tid = threadIdx.x;  // 128

  float mymax = -3.402823e38f;
  for (int s = tid; s < S_; s += 128) {
    const float* pp = partials + (size_t)(s * B_ + b) * 8;
    float v = 0.0f;
#pragma unroll
    for (int g = 0; g < 8; ++g) v += pp[g];
    sc[s] = v;
    mymax = fmaxf(mymax, v);
  }
  red[tid] = mymax;
  __syncthreads();
  for (int st = 64; st >= 1; st >>= 1) {
    if (tid < st) red[tid] = fmaxf(red[tid], red[tid + st]);
    __syncthreads();
  }
  const float mx = red[0];
  __syncthreads();

  float mysum = 0.0f;
  for (int s = tid; s < S_; s += 128) {
    const float e = expf(sc[s] - mx);
    sc[s] = e;
    mysum += e;
  }
  red[tid] = mysum;
  __syncthreads();
  for (int st = 64; st >= 1; st >>= 1) {
    if (tid < st) red[tid] += red[tid + st];
    __syncthreads();
  }
  const float inv = 1.0f / red[0];
  __syncthreads();
  for (int s = tid; s < S_; s += 128) attn_out[s * B_ + b] = sc[s] * inv;
}

// ===========================================================================
// K6: context[b,:] = sum_s attn[s,b]*enc[s,b,:]; p[b] = sigmoid(...).
// One block per b, 512 threads. enc read should hit L2 (resident from K4).
// ===========================================================================
__global__ void k_context_p(const float* __restrict__ enc,
                            const float* __restrict__ attn,
                            const float* __restrict__ x,
                            const float* __restrict__ pWh,
                            const float* __restrict__ pWs,
                            const float* __restrict__ pWx,
                            const float* __restrict__ pWx_b,
                            float* __restrict__ gs,
                            float* __restrict__ pbuf,
                            float* __restrict__ p_out) {
  __shared__ float a_s[S_];
  __shared__ float red[H_];
  const int b = blockIdx.x;
  const int h = threadIdx.x;  // 512
  for (int s = h; s < S_; s += H_) a_s[s] = attn[s * B_ + b];
  __syncthreads();

  float acc = 0.0f;
  for (int s = 0; s < S_; ++s)
    acc += a_s[s] * enc[(size_t)(s * B_ + b) * H_ + h];
  gs[(size_t)b * 1024 + H_ + h] = acc;           // context half of global_state

  const float hb = gs[(size_t)b * 1024 + h];
  red[h] = acc * pWh[h] + hb * pWs[h] + x[(size_t)b * H_ + h] * pWx[h];
  __syncthreads();
  for (int st = 256; st >= 1; st >>= 1) {
    if (h < st) red[h] += red[h + st];
    __syncthreads();
  }
  if (h == 0) {
    const float p = sigmoidf_(red[0] + pWx_b[0]);
    pbuf[b] = p;
    p_out[b] = p;
  }
}

// ===========================================================================
// K7: vocab projection (WMMA):  p * selu(gs(64x1024) @ out_W^T + out_b)
// Wave per n-tile (16 vocab rows), 4 M-tiles/wave -> out_W streamed ONCE.
// out_W loads + ext stores are NON-TEMPORAL: 205MB can't fit L2 anyway and
// RT policy would evict the L2-resident enc needed on the next graph replay.
// ===========================================================================
__global__ void k_vocab(const float* __restrict__ gs,
                        const float* __restrict__ out_W,
                        const float* __restrict__ out_b,
                        const float* __restrict__ pbuf,
                        float* __restrict__ ext) {
  const int gwave = (blockIdx.x * blockDim.x + threadIdx.x) >> 5;
  const int lane  = threadIdx.x & 31;
  if (gwave >= V_ / 16) return;  // 3125
  const int half = lane >> 4;
  const int l16  = lane & 15;
  const int vcol = gwave * 16 + l16;  // vocab index

  v8f acc[4];
#pragma unroll
  for (int mt = 0; mt < 4; ++mt) acc[mt] = v8zero();

  const float* wrow = out_W + (size_t)vcol * 1024;
  for (int k = 0; k < 1024; k += 4) {
    const int koff = k + half * 2;             // even -> 8B aligned
    v2f bm = __builtin_nontemporal_load((const v2f*)(wrow + koff));
#pragma unroll
    for (int mt = 0; mt < 4; ++mt) {
      const float* arow = gs + (size_t)(mt * 16 + l16) * 1024;
      v2f am = *(const v2f*)(arow + koff);
      acc[mt] = wmma_f32(am, bm, acc[mt]);
    }
  }

  const float ob = out_b[vcol];
  const float SC = 1.0507009873554805f;
  const float AL = 1.6732632423543772f;
#pragma unroll
  for (int mt = 0; mt < 4; ++mt) {
#pragma unroll
    for (int i = 0; i < 8; ++i) {
      const int b = mt * 16 + i + half * 8;
      const float logit = acc[mt][i] + ob;
      const float selu  = (logit > 0.0f) ? SC * logit : SC * AL * (expf(logit) - 1.0f);
      __builtin_nontemporal_store(pbuf[b] * selu, ext + (size_t)b * EXTW + vcol);
    }
  }
}

// ===========================================================================
// K8: OOV tail: ext[b, V+s] = (1-p[b]) * attn[s,b] * mask[s,b]
// ===========================================================================
__global__ void k_oov(const float* __restrict__ attn,
                      const int* __restrict__ mask,
                      const float* __restrict__ pbuf,
                      float* __restrict__ ext) {
  const int idx = blockIdx.x * blockDim.x + threadIdx.x;  // S*B = 25600
  if (idx >= S_ * B_) return;
  const int s = idx >> 6;
  const int b = idx & 63;
  __builtin_nontemporal_store((1.0f - pbuf[b]) * attn[idx] * (float)mask[idx],
                              ext + (size_t)b * EXTW + V_ + s);
}

// ===========================================================================
extern "C" void kernel_launch(void* const* d_in, const int* in_sizes, int n_in,
                              void* d_out, int out_size, void* d_ws, size_t ws_size,
                              hipStream_t stream) {
  (void)in_sizes; (void)n_in; (void)out_size; (void)ws_size;
  const float* x      = (const float*)d_in[0];
  const float* enc    = (const float*)d_in[1];
  const int*   mask   = (const int*)d_in[2];
  const float* h0     = (const float*)d_in[3];
  const float* W_ih   = (const float*)d_in[4];
  const float* W_hh   = (const float*)d_in[5];
  const float* b_ih   = (const float*)d_in[6];
  const float* b_hh   = (const float*)d_in[7];
  const float* out_W  = (const float*)d_in[8];
  const float* out_b  = (const float*)d_in[9];
  const float* vvec   = (const float*)d_in[10];
  const float* Wh     = (const float*)d_in[11];
  const float* Ws     = (const float*)d_in[12];
  const float* b_attn = (const float*)d_in[13];
  const float* pWh    = (const float*)d_in[14];
  const float* pWs    = (const float*)d_in[15];
  const float* pWx    = (const float*)d_in[16];
  const float* pWx_b  = (const float*)d_in[17];

  float* out = (float*)d_out;
  float* ws  = (float*)d_ws;

  float* gs    = ws + GS_OFF;
  float* gi    = ws + GI_OFF;
  float* gh    = ws + GH_OFF;
  float* x2    = ws + X2_OFF;
  float* parts = ws + PART_OFF;
  float* pbuf  = ws + P_OFF;

  // 1) GRU gate GEMMs (192 waves)
  k_gru_gates_gemm<<<48, 128, 0, stream>>>(x, h0, W_ih, W_hh, b_ih, b_hh, gi, gh);
  // 2) gate combine -> h
  k_gru_combine<<<(B_ * H_) / 256, 256, 0, stream>>>(gi, gh, h0, gs, out + HID_OFF);
  // 3) x2 = h @ Ws (32 waves)
  k_x2_gemm<<<8, 128, 0, stream>>>(gs, Ws, x2);
  // 4) attention score partials (12800 waves)
  k_attn_scores<<<1600, 256, 0, stream>>>(enc, Wh, x2, b_attn, vvec, parts);
  // 5) softmax over S
  k_softmax<<<B_, 128, 0, stream>>>(parts, out + ATTN_OFF);
  // 6) context + pointer prob
  k_context_p<<<B_, H_, 0, stream>>>(enc, out + ATTN_OFF, x, pWh, pWs, pWx, pWx_b,
                                     gs, pbuf, out + POUT_OFF);
  // 7) vocab projection (3125 waves; streams out_W exactly once, NT)
  k_vocab<<<(V_ / 16 + 3) / 4, 128, 0, stream>>>(gs, out_W, out_b, pbuf, out);
  // 8) OOV tail
  k_oov<<<(S_ * B_) / 256, 256, 0, stream>>>(out + ATTN_OFF, mask, pbuf, out);
}